// GraphConvLayer_74981539053710
// MI455X (gfx1250) — compile-verified
//
#include <hip/hip_runtime.h>

// Problem constants (match reference)
#define N_NODES 10000
#define N_EDGES 640000
#define D_FEAT  128
#define D_OUT   256
#define BN_EPS  1e-3f

// LDS column stride (floats) for the W slice: 128 + 4 pad -> conflict-free banks
#define LDS_COL_STRIDE 132

typedef __attribute__((ext_vector_type(2))) float v2f;
typedef __attribute__((ext_vector_type(8))) float v8f;
typedef __attribute__((ext_vector_type(4))) unsigned int u32x4;
typedef __attribute__((ext_vector_type(4))) int i32x4;
typedef __attribute__((ext_vector_type(8))) int i32x8;

// --- Tensor Data Mover availability / arity dispatch -----------------------
#if defined(__has_builtin)
#if __has_builtin(__builtin_amdgcn_tensor_load_to_lds)
#define HAVE_TDM 1
#if __has_include(<hip/amd_detail/amd_gfx1250_TDM.h>)
#define TDM_ARITY 6   /* amdgpu-toolchain (therock headers) -> 6-arg builtin */
#else
#define TDM_ARITY 5   /* ROCm 7.2 clang-22 -> 5-arg builtin */
#endif
#endif
#endif

// ---------------------------------------------------------------------------
// Kernel 1: zero the segment-sum accumulator and the counts buffer
// ---------------------------------------------------------------------------
__global__ void zero_ws_kernel(float* __restrict__ seg_sum, float* __restrict__ counts) {
    int idx = blockIdx.x * blockDim.x + threadIdx.x;
    int total = N_NODES * D_FEAT;
    if (idx < total) seg_sum[idx] = 0.0f;
    if (idx < N_NODES) counts[idx] = 0.0f;
}

// ---------------------------------------------------------------------------
// Kernel 2: wave-per-edge weighted scatter-add.
// One wave32 handles one edge: 32 lanes x float4 = 128 features.
// node_features (5.12 MB) + seg_sum (5.12 MB) are L2-resident on MI455X.
// ---------------------------------------------------------------------------
__global__ void edge_scatter_kernel(const float* __restrict__ feats,
                                    const int*   __restrict__ edges,   // [2, E]
                                    const float* __restrict__ ew,      // [E]
                                    float* __restrict__ seg_sum,       // [N, D]
                                    float* __restrict__ counts) {      // [N]
    const int lane = threadIdx.x & 31;
    const int wave = (blockIdx.x * blockDim.x + threadIdx.x) >> 5;
    const int nwaves = (gridDim.x * blockDim.x) >> 5;
    for (int e = wave; e < N_EDGES; e += nwaves) {
        const int dst = edges[e];            // edges[0][e] : segment id
        const int src = edges[N_EDGES + e];  // edges[1][e] : gather id
        const float w = ew[e];
        const float4 v = *(const float4*)(feats + src * D_FEAT + lane * 4);
        float* o = seg_sum + dst * D_FEAT + lane * 4;
        atomicAdd(o + 0, v.x * w);
        atomicAdd(o + 1, v.y * w);
        atomicAdd(o + 2, v.z * w);
        atomicAdd(o + 3, v.w * w);
        if (lane == 0) atomicAdd(counts + dst, 1.0f);
    }
}

// ---------------------------------------------------------------------------
// Kernel 3: segment-mean (empty segments -> 0) + inference BatchNorm
// ---------------------------------------------------------------------------
__global__ void mean_bn_kernel(const float* __restrict__ seg_sum,
                               const float* __restrict__ counts,
                               const float* __restrict__ gamma,
                               const float* __restrict__ beta,
                               const float* __restrict__ mmean,
                               const float* __restrict__ mvar,
                               float* __restrict__ bn) {
    int idx = blockIdx.x * blockDim.x + threadIdx.x;
    if (idx >= N_NODES * D_FEAT) return;
    const int f = idx & (D_FEAT - 1);
    const int n = idx >> 7;  // D_FEAT == 128
    const float c = counts[n];
    const float agg = (c > 0.0f) ? (seg_sum[idx] / fmaxf(c, 1.0f)) : 0.0f;
    const float scale = __frsqrt_rn(mvar[f] + BN_EPS) * gamma[f];
    bn[idx] = (agg - mmean[f]) * scale + beta[f];
}

// ---------------------------------------------------------------------------
// Kernel 3b: transpose W [128 x 256] -> Wt [256 x 128] (one-time, 128 KB)
// Column-major W lets the GEMM fetch B k-pairs as single 8-byte LDS loads.
// ---------------------------------------------------------------------------
__global__ void transpose_w_kernel(const float* __restrict__ W, float* __restrict__ Wt) {
    int idx = blockIdx.x * blockDim.x + threadIdx.x;
    if (idx >= D_FEAT * D_OUT) return;
    const int k = idx >> 8;          // 0..127
    const int n = idx & (D_OUT - 1); // 0..255
    Wt[n * D_FEAT + k] = W[k * D_OUT + n];
}

// ---------------------------------------------------------------------------
// Kernel 4: WMMA GEMM  out = relu(bn @ W + b)   (full fp32 matrix path)
//   block = 256 threads = 8 waves; block tile = 128 rows x 64 cols
//   wave tile = 16 rows x 64 cols (4 accumulator sets, A regs reused 4x)
//   Wt slice (64 cols x 128 k, column-major, stride 132) staged in LDS —
//   via the Tensor Data Mover (TDM) when available (padded 2D tile DMA),
//   else a cooperative vector-load fallback.
//
// V_WMMA_F32_16X16X4_F32 wave32 layouts (ISA 7.12.2):
//   A 16x4 : lane l, vgpr v -> A[M = l&15, K = 2*(l>>4) + v]
//   B 4x16 : lane l, vgpr v -> B[K = 2*(l>>4) + v, N = l&15]
//   C/D    : vgpr r, lane l -> D[M = r + ((l>>4)*8), N = l&15]
// ---------------------------------------------------------------------------
__global__ void gemm_wmma_kernel(const float* __restrict__ A,   // bn [N x 128]
                                 const float* __restrict__ Wt,  // [256 x 128] (W^T)
                                 const float* __restrict__ bias,// [256]
                                 float* __restrict__ out) {     // [N x 256]
    __shared__ float lW[64 * LDS_COL_STRIDE];  // ~33 KB, column-major W slice

    const int n0 = blockIdx.y * 64;
    const int m_block = blockIdx.x * 128;
    const int tid = threadIdx.x;

#if defined(HAVE_TDM)
    // --- TDM: DMA the 2D tile Wt[n0 : n0+64][0:128] into LDS, padding the
    //     column stride from 128 to 132 dwords (pad_interval=6 -> every 128
    //     dwords, pad_amount=3 -> 4 dwords). One descriptor, issued by wave 0.
    if (tid < 32) {
        const unsigned long long ga =
            (unsigned long long)(const void*)(Wt + (size_t)n0 * D_FEAT);
        const unsigned lds_addr = (unsigned)(unsigned long long)(void*)&lW[0];
        u32x4 g0;
        g0.x = 1u;                                            // count = 1 valid D#
        g0.y = lds_addr;                                      // lds_addr [63:32]
        g0.z = (unsigned)(ga & 0xFFFFFFFFull);                // global_addr lo
        g0.w = (unsigned)((ga >> 32) & 0x01FFFFFFull)         // global_addr hi
             | (2u << 30);                                    // type = 2 (image)
        i32x8 g1;
        g1[0] = (int)((2u << 16)      // data_size = 2 -> 4-byte elements
                    | (1u << 20)      // pad_enable
                    | (6u << 22)      // pad_interval: every 128 dwords
                    | (3u << 25));    // pad_amount: 4 dwords
        g1[1] = (int)(((unsigned)D_FEAT) << 16);  // tensor_dim0 = 128 (lo16)
        g1[2] = (int)(64u << 16);                 // dim0 hi = 0 | tensor_dim1 = 64
        g1[3] = (int)(((unsigned)D_FEAT) << 16);  // dim1 hi = 0 | tile_dim0 = 128
        g1[4] = (int)64;                          // tile_dim1 = 64 | tile_dim2 = 0
        g1[5] = (int)D_FEAT;                      // tensor_dim0_stride = 128
        g1[6] = 0;
        g1[7] = 0;
        i32x4 z4 = {0, 0, 0, 0};
#if TDM_ARITY == 6
        i32x8 z8 = {0, 0, 0, 0, 0, 0, 0, 0};
        __builtin_amdgcn_tensor_load_to_lds(g0, g1, z4, z4, z8, 0);
#else
        __builtin_amdgcn_tensor_load_to_lds(g0, g1, z4, z4, 0);
#endif
#if __has_builtin(__builtin_amdgcn_s_wait_tensorcnt)
        __builtin_amdgcn_s_wait_tensorcnt(0);
#else
        asm volatile("s_wait_tensorcnt 0x0" ::: "memory");
#endif
    }
#else
    // --- Fallback: cooperative b128 staged load, column-major with pad
    for (int i = tid; i < 64 * 32; i += 256) {
        const int n  = i >> 5;   // column 0..63
        const int c4 = i & 31;   // float4 chunk along k
        const float4 v = *(const float4*)(Wt + (size_t)(n0 + n) * D_FEAT + c4 * 4);
        *(float4*)(lW + n * LDS_COL_STRIDE + c4 * 4) = v;
    }
#endif
    __syncthreads();

    const int wave = tid >> 5;            // 0..7 -> M sub-tile
    const int lane = tid & 31;
    const int row0 = m_block + wave * 16;
    const int a_row = row0 + (lane & 15);
    const int a_row_c = a_row < N_NODES ? a_row : (N_NODES - 1);  // clamp OOB loads
    const float* __restrict__ Arow = A + (size_t)a_row_c * D_FEAT;
    const int khi = (lane >> 4) * 2;      // 0 or 2
    const int bn_col = lane & 15;

    __builtin_prefetch(Arow, 0, 1);       // global_prefetch: pull A row toward L0/L2

    // Per-tile LDS column bases (pair loads are 8B-aligned: khi, k even)
    const float* __restrict__ c0 = lW + (0 * 16 + bn_col) * LDS_COL_STRIDE + khi;
    const float* __restrict__ c1 = lW + (1 * 16 + bn_col) * LDS_COL_STRIDE + khi;
    const float* __restrict__ c2 = lW + (2 * 16 + bn_col) * LDS_COL_STRIDE + khi;
    const float* __restrict__ c3 = lW + (3 * 16 + bn_col) * LDS_COL_STRIDE + khi;

    v8f acc0 = {}, acc1 = {}, acc2 = {}, acc3 = {};

    #pragma unroll
    for (int k = 0; k < D_FEAT; k += 4) {
        const v2f a = *(const v2f*)(Arow + k + khi);   // global b64
        const v2f b0 = *(const v2f*)(c0 + k);          // ds b64, no repack
        const v2f b1 = *(const v2f*)(c1 + k);
        const v2f b2 = *(const v2f*)(c2 + k);
        const v2f b3 = *(const v2f*)(c3 + k);
        acc0 = __builtin_amdgcn_wmma_f32_16x16x4_f32(false, a, false, b0, (short)0, acc0, false, false);
        acc1 = __builtin_amdgcn_wmma_f32_16x16x4_f32(false, a, false, b1, (short)0, acc1, false, false);
        acc2 = __builtin_amdgcn_wmma_f32_16x16x4_f32(false, a, false, b2, (short)0, acc2, false, false);
        acc3 = __builtin_amdgcn_wmma_f32_16x16x4_f32(false, a, false, b3, (short)0, acc3, false, false);
    }

    // epilogue: bias + relu + store (guard M bounds)
    const int m_base = row0 + ((lane >> 4) * 8);
    v8f accs[4] = {acc0, acc1, acc2, acc3};
    #pragma unroll
    for (int t = 0; t < 4; ++t) {
        const int n = n0 + t * 16 + bn_col;
        const float bv = bias[n];
        #pragma unroll
        for (int r = 0; r < 8; ++r) {
            const int m = m_base + r;
            if (m < N_NODES) {
                const float v = accs[t][r] + bv;
                out[(size_t)m * D_OUT + n] = v > 0.0f ? v : 0.0f;
            }
        }
    }
}

// ---------------------------------------------------------------------------
extern "C" void kernel_launch(void* const* d_in, const int* in_sizes, int n_in,
                              void* d_out, int out_size, void* d_ws, size_t ws_size,
                              hipStream_t stream) {
    const float* node_features = (const float*)d_in[0];
    const int*   edges         = (const int*)  d_in[1];
    const float* edge_weights  = (const float*)d_in[2];
    const float* gamma         = (const float*)d_in[3];
    const float* beta          = (const float*)d_in[4];
    const float* moving_mean   = (const float*)d_in[5];
    const float* moving_var    = (const float*)d_in[6];
    const float* W             = (const float*)d_in[7];
    const float* b             = (const float*)d_in[8];
    float* out = (float*)d_out;

    // workspace layout (floats): seg_sum | counts | bn | Wt
    float* seg_sum = (float*)d_ws;                        // N_NODES * D_FEAT
    float* counts  = seg_sum + N_NODES * D_FEAT;          // N_NODES
    float* bn      = counts + N_NODES;                    // N_NODES * D_FEAT
    float* Wt      = bn + N_NODES * D_FEAT;               // D_OUT * D_FEAT

    // 1) zero accumulators
    {
        const int total = N_NODES * D_FEAT;
        zero_ws_kernel<<<(total + 255) / 256, 256, 0, stream>>>(seg_sum, counts);
    }
    // 2) wave-per-edge scatter (grid-stride over 640K edges)
    edge_scatter_kernel<<<2048, 256, 0, stream>>>(node_features, edges, edge_weights,
                                                  seg_sum, counts);
    // 3) segment-mean + batchnorm, and W transpose (independent, same stream)
    {
        const int total = N_NODES * D_FEAT;
        mean_bn_kernel<<<(total + 255) / 256, 256, 0, stream>>>(
            seg_sum, counts, gamma, beta, moving_mean, moving_var, bn);
        transpose_w_kernel<<<(D_FEAT * D_OUT + 255) / 256, 256, 0, stream>>>(W, Wt);
    }
    // 4) WMMA f32 GEMM + bias + relu
    {
        dim3 grid((N_NODES + 127) / 128, D_OUT / 64);
        gemm_wmma_kernel<<<grid, 256, 0, stream>>>(bn, Wt, b, out);
    }
}